// SIMPLESampler_15934328668770
// MI455X (gfx1250) — compile-verified
//
#include <hip/hip_runtime.h>
#include <hip/hip_bf16.h>
#include <stdint.h>

// Problem dims (fixed by reference): B=512 rows, N=8192 items, K=1.
#define BDIM 512
#define NDIM 8192
#define ROWS_PER_WG 32      // one wave32 per workgroup, lane = row
#define CHUNK 64            // items staged per TDM tile
#define NCHUNK (NDIM / CHUNK)
#define LDS_PITCH (CHUNK + 1)                 // 64 data dwords + 1 pad dword (TDM pad_interval=5)
#define TILE_DW (ROWS_PER_WG * LDS_PITCH)     // 2080 dwords
#define TILE_BYTES (TILE_DW * 4)              // 8320 B; 4 tiles = 33280 B LDS

#if defined(__HIP_DEVICE_COMPILE__) && \
    __has_builtin(__builtin_amdgcn_tensor_load_to_lds) && \
    __has_builtin(__builtin_amdgcn_s_wait_tensorcnt)
#define HAVE_TDM 1
#else
#define HAVE_TDM 0
#endif

// ---------- numerics: replicate the JAX formulas op-for-op ----------
__device__ __forceinline__ float log1mexp_f(float x) {
  // log(1 - exp(x)) for x <= 0 (Maechler 2012), matching reference branch
  return (x > -0.6931471805599453f) ? logf(-expm1f(x)) : log1pf(-expf(x));
}

// One DP step + Bernoulli candidate test for element e.
// S,T are the sequential log-space accumulators (bit-faithful order).
__device__ __forceinline__ void step_update(float x, float u, int e,
                                            float& S, float& T, int& selIdx) {
  // logp = min(log_sigmoid(x), -1e-7); log_sigmoid = min(x,0) - log1p(exp(-|x|))
  float lsig = fminf(x, 0.0f) - log1pf(expf(-fabsf(x)));
  float lp = fminf(lsig, -1e-7f);
  float lq = log1mexp_f(lp);
  // T' = logaddexp_c(S + lp, T + lq) ; S' = S + lq
  float x1 = S + lp;
  float x2 = T + lq;                       // -inf + finite = -inf handled naturally
  float delta = (x1 == x2) ? 0.0f : (x1 - x2);
  T = fmaxf(x1, x2) + log1pf(expf(-fabsf(delta)));
  S = S + lq;
  // reverse-pass candidate (j == K+1 column): fires -> this e could be selected
  float p = fminf(x1 - T, 0.0f);
  float q = log1mexp_f(p);
  float zz = p - q;
  float prob = 1.0f / (1.0f + expf(-zz));  // sigmoid(p - q)
  if (u < prob) selIdx = e;                // final value = largest firing e
}

// ---------- Tensor Data Mover staging ----------
#if HAVE_TDM
typedef unsigned int v4u_t __attribute__((ext_vector_type(4)));
typedef int          v8i_t __attribute__((ext_vector_type(8)));
typedef int          v4i_t __attribute__((ext_vector_type(4)));

// Load a 2-D tile [ROWS_PER_WG rows x CHUNK elems] of f32 from a row-major
// [BDIM x NDIM] tensor into LDS at byte offset lds_off, padding 1 dword per
// 64-dword row (pad_interval=5, pad_amount=0 => pitch 65 => bank-conflict-free).
__device__ __forceinline__ void tdm_load_tile(const float* gtile, uint32_t lds_off) {
  uint64_t ga = (uint64_t)(uintptr_t)gtile;
  v4u_t g0;
  g0.x = 1u;                                            // count=1 valid descriptor
  g0.y = lds_off;                                       // lds_addr (bytes)
  g0.z = (uint32_t)ga;                                  // global_addr[31:0]
  g0.w = (uint32_t)((ga >> 32) & 0x1FFFFFFull)          // global_addr[56:32]
       | (2u << 30);                                    // type = 2 ("image")
  v8i_t g1;
  g1[0] = (int)((2u << 16)                              // data_size = 4 bytes
              | (1u << 20)                              // pad_enable
              | (5u << 22));                            // pad_interval: 2^(5+1)=64 dwords
                                                        // pad_amount = 0 -> 1 dword
  g1[1] = (int)(((uint32_t)NDIM & 0xFFFFu) << 16);      // tensor_dim0 lo16 (bits 79:48)
  g1[2] = (int)((((uint32_t)NDIM >> 16) & 0xFFFFu)      // tensor_dim0 hi16
              | (((uint32_t)BDIM & 0xFFFFu) << 16));    // tensor_dim1 lo16
  g1[3] = (int)((((uint32_t)BDIM >> 16) & 0xFFFFu)      // tensor_dim1 hi16
              | ((uint32_t)CHUNK << 16));               // tile_dim0 = CHUNK
  g1[4] = (int)(ROWS_PER_WG);                           // tile_dim1 = 32 rows, tile_dim2 = 0
  g1[5] = (int)NDIM;                                    // tensor_dim0_stride lo32
  g1[6] = 0;                                            // stride0 hi16 | stride1 lo16
  g1[7] = 0;                                            // stride1 hi32 (unused, 2-D tile)
  v4i_t z4 = {0, 0, 0, 0};
  v8i_t z8 = {0, 0, 0, 0, 0, 0, 0, 0};
  // clang-23 / therock-10.0 arity: (g0, g1, g2, g3, g4, cpol)
  __builtin_amdgcn_tensor_load_to_lds(g0, g1, z4, z4, z8, 0);
}
#endif

// ---------- kernel 1: sequential DP scan + sampling, one wave32 per 32 rows ----------
__global__ void __launch_bounds__(ROWS_PER_WG)
scan_sample_kernel(const float* __restrict__ logits,
                   const float* __restrict__ noise,
                   int* __restrict__ sel) {
  extern __shared__ float smem[];   // 4 tiles: [buf0:lg][buf0:u][buf1:lg][buf1:u]
  const int lane = threadIdx.x;
  const int row  = blockIdx.x * ROWS_PER_WG + lane;

  float S = 0.0f;
  float T = -INFINITY;
  int selIdx = -1;

#if HAVE_TDM
  const float* grow = logits + (size_t)blockIdx.x * ROWS_PER_WG * NDIM;
  const float* nrow = noise  + (size_t)blockIdx.x * ROWS_PER_WG * NDIM;
  // prime chunk 0
  tdm_load_tile(grow, 0u * TILE_BYTES);
  tdm_load_tile(nrow, 1u * TILE_BYTES);
  for (int c = 0; c < NCHUNK; ++c) {
    const int par = c & 1;
    if (c + 1 < NCHUNK) {
      const int np = par ^ 1;
      tdm_load_tile(grow + (size_t)(c + 1) * CHUNK, (uint32_t)(np * 2 + 0) * TILE_BYTES);
      tdm_load_tile(nrow + (size_t)(c + 1) * CHUNK, (uint32_t)(np * 2 + 1) * TILE_BYTES);
      __builtin_amdgcn_s_wait_tensorcnt(2);   // chunk c's two tiles complete (in-order)
    } else {
      __builtin_amdgcn_s_wait_tensorcnt(0);
    }
    __asm__ volatile("" ::: "memory");        // keep ds reads after the tensorcnt wait
    const float* lb = smem + (size_t)(par * 2 + 0) * TILE_DW + lane * LDS_PITCH;
    const float* ub = smem + (size_t)(par * 2 + 1) * TILE_DW + lane * LDS_PITCH;
#pragma unroll 8
    for (int t = 0; t < CHUNK; ++t) {
      step_update(lb[t], ub[t], c * CHUNK + t, S, T, selIdx);
    }
    __asm__ volatile("" ::: "memory");        // finish reads before buffer is re-filled
  }
#else
  // Fallback: direct (uncoalesced) global reads, same math.
  const float* lrow = logits + (size_t)row * NDIM;
  const float* urow = noise  + (size_t)row * NDIM;
  for (int e = 0; e < NDIM; ++e) {
    __builtin_prefetch(lrow + e + 64, 0, 0);
    step_update(lrow[e], urow[e], e, S, T, selIdx);
  }
#endif
  sel[row] = selIdx;
}

// ---------- kernel 2: scatter the one-hot result at full bandwidth ----------
__global__ void write_out_kernel(const int* __restrict__ sel, float* __restrict__ out) {
  int idx = blockIdx.x * blockDim.x + threadIdx.x;
  int b = idx >> 13;          // / NDIM (8192)
  int e = idx & (NDIM - 1);
  out[idx] = (sel[b] == e) ? 1.0f : 0.0f;
}

extern "C" void kernel_launch(void* const* d_in, const int* in_sizes, int n_in,
                              void* d_out, int out_size, void* d_ws, size_t ws_size,
                              hipStream_t stream) {
  (void)in_sizes; (void)n_in; (void)out_size; (void)ws_size;
  const float* logits = (const float*)d_in[0];
  const float* noise  = (const float*)d_in[1];
  int* sel = (int*)d_ws;   // B ints of scratch

  scan_sample_kernel<<<BDIM / ROWS_PER_WG, ROWS_PER_WG, 4 * TILE_BYTES, stream>>>(
      logits, noise, sel);
  write_out_kernel<<<(BDIM * NDIM) / 256, 256, 0, stream>>>(sel, (float*)d_out);
}